// RetinaNetLoss_50852412785050
// MI455X (gfx1250) — compile-verified
//
#include <hip/hip_runtime.h>
#include <hip/hip_bf16.h>
#include <math.h>

typedef __attribute__((ext_vector_type(2))) float v2f;
typedef __attribute__((ext_vector_type(4))) float f4;
typedef __attribute__((ext_vector_type(8))) float v8f;

#define APB  256     // anchors per block == blockDim.x
#define NCLS 80
#define NC4  (NCLS / 4)   // 20 float4 per anchor

__device__ __forceinline__ float focal_term(float p, bool ispos) {
    // clip to [EPS, 1-EPS]
    p = fminf(fmaxf(p, 1e-4f), 1.0f - 1e-4f);
    // pe = prob of the "wrong" outcome; bce = -log(1 - pe) in BOTH branches:
    //   ispos:  -log(p)    = -log(1 - (1-p))
    //   !ispos: -log1p(-p) = -log(1 - p)
    float pe = ispos ? (1.0f - p) : p;
    float w  = ispos ? 0.25f : 0.75f;          // alpha / (1-alpha)
    return w * pe * pe * (-__logf(1.0f - pe)); // gamma == 2
}

__global__ __launch_bounds__(APB) void retina_partial_kernel(
    const f4* __restrict__ reg,     // [B*A] float4
    const f4* __restrict__ cls4,    // [B*A*NC4] float4
    const float* __restrict__ ann,  // [B*M*5]
    const f4* __restrict__ anc4,    // [B*A] float4
    const int* __restrict__ img,    // [2] = {H, W}
    float* __restrict__ ws,         // [3 * nblocks_total] partials
    int A, int M, int nblocks_total)
{
    __shared__ float s_ann[5 * 64];
    __shared__ int   s_pcls[APB];
    __shared__ float s_valid[APB];
    __shared__ float s_red[24];

    const int tid  = threadIdx.x;
    const int b    = blockIdx.y;
    const int ablk = blockIdx.x;
    const int a    = ablk * APB + tid;

    if (tid < M * 5) s_ann[tid] = ann[b * M * 5 + tid];
    __syncthreads();

    float regsum = 0.0f, nposf = 0.0f, validf = 0.0f;
    int   pcls   = -1;

    if (a < A) {
        f4 an = anc4[(size_t)b * A + a];
        float aw = an.z - an.x, ah = an.w - an.y;
        float area_a = aw * ah;
        float cx = 0.5f * (an.x + an.z), cy = 0.5f * (an.y + an.w);
        bool outside = (cx >= (float)img[1]) || (cy >= (float)img[0]);

        float best_iou = -1.0f; int best = 0;
        #pragma unroll 4
        for (int m = 0; m < M; ++m) {
            float bx1 = s_ann[m * 5 + 0], by1 = s_ann[m * 5 + 1];
            float bx2 = s_ann[m * 5 + 2], by2 = s_ann[m * 5 + 3];
            float iw = fmaxf(fminf(an.z, bx2) - fmaxf(an.x, bx1), 0.0f);
            float ih = fmaxf(fminf(an.w, by2) - fmaxf(an.y, by1), 0.0f);
            float inter  = iw * ih;
            float area_b = (bx2 - bx1) * (by2 - by1);
            float uni    = fmaxf(area_a + area_b - inter, 1e-8f);
            float iou    = inter / uni;
            if (iou > best_iou) { best_iou = iou; best = m; }
        }
        bool pos  = best_iou >= 0.5f;
        bool ign  = (best_iou > 0.4f) && !pos;
        bool spos = pos && !outside;                          // states == 1
        validf = (!outside && (pos || !ign)) ? 1.0f : 0.0f;   // states != -1
        pcls   = pos ? (int)s_ann[best * 5 + 4] : -1;
        nposf  = spos ? 1.0f : 0.0f;

        // ---- regression (smooth L1, sigma=3 -> sigma^2=9; targets scaled /0.2) ----
        float g0 = s_ann[best * 5 + 0], g1 = s_ann[best * 5 + 1];
        float g2 = s_ann[best * 5 + 2], g3 = s_ann[best * 5 + 3];
        float iaw = 5.0f / aw, iah = 5.0f / ah;
        float t0 = (g0 - an.x) * iaw, t1 = (g1 - an.y) * iah;
        float t2 = (g2 - an.z) * iaw, t3 = (g3 - an.w) * iah;
        f4 r = __builtin_nontemporal_load(&reg[(size_t)b * A + a]);
        float d0 = fabsf(r.x - t0), d1 = fabsf(r.y - t1);
        float d2 = fabsf(r.z - t2), d3 = fabsf(r.w - t3);
        const float ss = 9.0f, iss = 1.0f / 9.0f;
        float s0 = d0 < iss ? 0.5f * ss * d0 * d0 : d0 - 0.5f * iss;
        float s1 = d1 < iss ? 0.5f * ss * d1 * d1 : d1 - 0.5f * iss;
        float s2 = d2 < iss ? 0.5f * ss * d2 * d2 : d2 - 0.5f * iss;
        float s3 = d3 < iss ? 0.5f * ss * d3 * d3 : d3 - 0.5f * iss;
        regsum = nposf * (s0 + s1 + s2 + s3);
    }

    s_pcls[tid]  = pcls;
    s_valid[tid] = validf;
    __syncthreads();

    // ---- classification: coalesced class-major sweep over this block's anchors ----
    float clssum = 0.0f;
    int nanch = min(APB, A - ablk * APB);
    int n4    = nanch * NC4;
    const f4* cbase = cls4 + ((size_t)b * A + (size_t)ablk * APB) * NC4;
    for (int j = tid; j < APB * NC4; j += APB) {
        // speculative prefetch of next tile; OOB translation failures are dropped
        __builtin_prefetch(&cbase[j + APB], 0, 0);
        if (j < n4) {
            f4 p = __builtin_nontemporal_load(&cbase[j]);
            unsigned ju = (unsigned)j;
            int al = (int)(ju / (unsigned)NC4);
            int c0 = (int)(ju % (unsigned)NC4) * 4;
            int pc = s_pcls[al];
            float v = s_valid[al];
            float t = focal_term(p.x, c0     == pc)
                    + focal_term(p.y, c0 + 1 == pc)
                    + focal_term(p.z, c0 + 2 == pc)
                    + focal_term(p.w, c0 + 3 == pc);
            clssum += v * t;
        }
    }

    // ---- block reduction (wave32 shuffles + LDS across 8 waves) ----
    float vals[3] = { clssum, regsum, nposf };
    int lane = tid & 31, wid = tid >> 5;
    #pragma unroll
    for (int c = 0; c < 3; ++c) {
        float v = vals[c];
        for (int off = 16; off > 0; off >>= 1) v += __shfl_down(v, off, 32);
        if (lane == 0) s_red[c * 8 + wid] = v;
    }
    __syncthreads();
    if (tid == 0) {
        int id = blockIdx.y * gridDim.x + blockIdx.x;
        #pragma unroll
        for (int c = 0; c < 3; ++c) {
            float s = 0.0f;
            for (int w = 0; w < 8; ++w) s += s_red[c * 8 + w];
            ws[(size_t)c * nblocks_total + id] = s;
        }
    }
}

// Final deterministic reduction of block partials using V_WMMA_F32_16X16X4_F32:
// A = 64 partials striped over the wave (2 per lane, one b64 load), B = ones(4x16),
// C accumulates across chunks. Three channels interleaved -> 3 independent WMMA
// accumulator chains per iteration (no RAW between them, latency pipelines).
__global__ void retina_finalize_kernel(const float* __restrict__ ws, int n,
                                       float* __restrict__ out)
{
    int lane = threadIdx.x;   // launched with exactly 32 threads (one wave32)
    const float* b0 = ws;
    const float* b1 = ws + (size_t)n;
    const float* b2 = ws + (size_t)2 * n;
    const v2f* p0 = (const v2f*)b0;
    const v2f* p1 = (const v2f*)b1;
    const v2f* p2 = (const v2f*)b2;

    v2f ones = { 1.0f, 1.0f };
    v8f acc0 = {}, acc1 = {}, acc2 = {};

    int full = n / 64;        // full 64-element chunks: unguarded b64 loads
    for (int k = 0; k < full; ++k) {
        v2f a0 = p0[k * 32 + lane];
        v2f a1 = p1[k * 32 + lane];
        v2f a2 = p2[k * 32 + lane];
        acc0 = __builtin_amdgcn_wmma_f32_16x16x4_f32(false, a0, false, ones,
                                                     (short)0, acc0, false, false);
        acc1 = __builtin_amdgcn_wmma_f32_16x16x4_f32(false, a1, false, ones,
                                                     (short)0, acc1, false, false);
        acc2 = __builtin_amdgcn_wmma_f32_16x16x4_f32(false, a2, false, ones,
                                                     (short)0, acc2, false, false);
    }
    // guarded tail chunk (contributes zeros if n % 64 == 0)
    {
        int i0 = full * 64 + lane * 2;
        v2f a0, a1, a2;
        a0[0] = (i0 < n) ? b0[i0] : 0.0f;  a0[1] = (i0 + 1 < n) ? b0[i0 + 1] : 0.0f;
        a1[0] = (i0 < n) ? b1[i0] : 0.0f;  a1[1] = (i0 + 1 < n) ? b1[i0 + 1] : 0.0f;
        a2[0] = (i0 < n) ? b2[i0] : 0.0f;  a2[1] = (i0 + 1 < n) ? b2[i0 + 1] : 0.0f;
        acc0 = __builtin_amdgcn_wmma_f32_16x16x4_f32(false, a0, false, ones,
                                                     (short)0, acc0, false, false);
        acc1 = __builtin_amdgcn_wmma_f32_16x16x4_f32(false, a1, false, ones,
                                                     (short)0, acc1, false, false);
        acc2 = __builtin_amdgcn_wmma_f32_16x16x4_f32(false, a2, false, ones,
                                                     (short)0, acc2, false, false);
    }

    // D rows replicated across N; rows 0-7 live in lanes 0-15, rows 8-15 in 16-31.
    float s0 = acc0[0]+acc0[1]+acc0[2]+acc0[3]+acc0[4]+acc0[5]+acc0[6]+acc0[7];
    float s1 = acc1[0]+acc1[1]+acc1[2]+acc1[3]+acc1[4]+acc1[5]+acc1[6]+acc1[7];
    float s2 = acc2[0]+acc2[1]+acc2[2]+acc2[3]+acc2[4]+acc2[5]+acc2[6]+acc2[7];
    float cls = __shfl(s0, 0, 32) + __shfl(s0, 16, 32);
    float rg  = __shfl(s1, 0, 32) + __shfl(s1, 16, 32);
    float np  = __shfl(s2, 0, 32) + __shfl(s2, 16, 32);
    if (lane == 0)
        out[0] = (cls + rg) / fmaxf(np, 1.0f);
}

extern "C" void kernel_launch(void* const* d_in, const int* in_sizes, int n_in,
                              void* d_out, int out_size, void* d_ws, size_t ws_size,
                              hipStream_t stream) {
    const f4*    reg  = (const f4*)d_in[0];     // output_regression  [B,A,4]
    const f4*    cls4 = (const f4*)d_in[1];     // output_classification [B,A,80]
    const float* ann  = (const float*)d_in[2];  // batch_annotations  [B,M,5]
    const f4*    anc4 = (const f4*)d_in[3];     // anchors            [B,A,4]
    const int*   img  = (const int*)d_in[4];    // image_shape        [2]

    const int M = 32;                            // per reference
    int B = in_sizes[2] / (M * 5);
    int A = in_sizes[0] / (4 * B);
    int nb  = (A + APB - 1) / APB;
    int nbt = B * nb;                            // total partial blocks (~3752)

    float* ws = (float*)d_ws;                    // 3 * nbt floats (~45 KB)
    dim3 grid(nb, B);
    retina_partial_kernel<<<grid, APB, 0, stream>>>(reg, cls4, ann, anc4, img,
                                                    ws, A, M, nbt);
    retina_finalize_kernel<<<1, 32, 0, stream>>>(ws, nbt, (float*)d_out);
}